// GNN_979252543708
// MI455X (gfx1250) — compile-verified
//
#include <hip/hip_runtime.h>

#define N_NODES  100000
#define N_EDGES  2500000
#define N_GRAPHS 256
#define IN_FEATS 6
#define HIDDEN   32
#define EMB      64

typedef __attribute__((ext_vector_type(2))) float v2f;
typedef __attribute__((ext_vector_type(8))) float v8f;

// ---------------------------------------------------------------- zeroing
__global__ void zero_f32(float* __restrict__ p, int n) {
    int i = blockIdx.x * blockDim.x + threadIdx.x;
    if (i < n) p[i] = 0.0f;
}

// ------------------------------------------- layer-1 scatter: deg + agg1
__global__ void scatter1(const float* __restrict__ x,
                         const int* __restrict__ src,
                         const int* __restrict__ dst,
                         float* __restrict__ deg,
                         float* __restrict__ agg1) {
    int e = blockIdx.x * blockDim.x + threadIdx.x;
    if (e >= N_EDGES) return;
    int s = src[e];
    int d = dst[e];
    atomicAdd(&deg[d], 1.0f);
    const float* xs = x + (long)s * IN_FEATS;
    float* ad = agg1 + (long)d * IN_FEATS;
#pragma unroll
    for (int j = 0; j < IN_FEATS; ++j) atomicAdd(&ad[j], xs[j]);
}

// -------------------- layer 1: h1 = relu(((agg1+x)/(deg+1)) @ W1 + b1)
// one wave per 16-node tile; M=16, N=32 (2 tiles), K=6 padded to 8 (2 k-steps).
// Pad lanes use a clamped (safe) address and a zeroed A multiplier — no
// divergent branches, EXEC stays all-ones for every WMMA.
__global__ void layer1_wmma(const float* __restrict__ x,
                            const float* __restrict__ agg1,
                            const float* __restrict__ deg,
                            const float* __restrict__ W1,
                            const float* __restrict__ bias1,
                            float* __restrict__ h1) {
    const int lane = threadIdx.x & 31;
    const int wave = threadIdx.x >> 5;
    const int tile = blockIdx.x * (blockDim.x >> 5) + wave;
    const int NT = N_NODES / 16;              // 6250, exact
    if (tile >= NT) return;                   // wave-uniform exit

    const int row  = lane & 15;               // M for A-frag, N-within-tile for B/C/D
    const int kh   = lane >> 4;               // which K half-pair this lane holds
    const int node = tile * 16 + row;
    const float inv = 1.0f / (deg[node] + 1.0f);
    const long base = (long)node * IN_FEATS;

    v8f c0 = {};  // N = 0..15
    v8f c1 = {};  // N = 16..31
#pragma unroll
    for (int kk = 0; kk < 2; ++kk) {
        const int k0 = kk * 4 + kh * 2;       // lane-varying K base: 0,2,4,6
        // k0 is even, so k0<6 <=> k0+1<6: a single pad predicate per lane
        const int   kc0 = (k0     < IN_FEATS) ? k0     : 0;  // clamped safe index
        const int   kc1 = (k0 + 1 < IN_FEATS) ? k0 + 1 : 0;
        const float mz  = (k0 < IN_FEATS) ? inv : 0.0f;      // zero pad lanes via A
        v2f a;
        a.x = (agg1[base + kc0] + x[base + kc0]) * mz;
        a.y = (agg1[base + kc1] + x[base + kc1]) * mz;
        v2f b0, b1f;                           // B needs no zeroing: A==0 on pad
        b0.x  = W1[kc0 * HIDDEN + row];
        b0.y  = W1[kc1 * HIDDEN + row];
        b1f.x = W1[kc0 * HIDDEN + 16 + row];
        b1f.y = W1[kc1 * HIDDEN + 16 + row];
        c0 = __builtin_amdgcn_wmma_f32_16x16x4_f32(false, a, false, b0,  (short)0, c0, false, false);
        c1 = __builtin_amdgcn_wmma_f32_16x16x4_f32(false, a, false, b1f, (short)0, c1, false, false);
    }

    // D layout: VGPR r holds M = r + 8*kh, N = ntile*16 + (lane&15)
#pragma unroll
    for (int r = 0; r < 8; ++r) {
        const int m    = r + 8 * kh;
        const int orow = tile * 16 + m;
        float v0 = c0[r] + bias1[row];
        float v1 = c1[r] + bias1[16 + row];
        h1[(long)orow * HIDDEN + row]      = v0 > 0.0f ? v0 : 0.0f;
        h1[(long)orow * HIDDEN + 16 + row] = v1 > 0.0f ? v1 : 0.0f;
    }
}

// ------------------------------- layer-2 scatter: agg2 (8 threads/edge)
__global__ void scatter2(const float* __restrict__ h1,
                         const int* __restrict__ src,
                         const int* __restrict__ dst,
                         float* __restrict__ agg2) {
    int gid = blockIdx.x * blockDim.x + threadIdx.x;
    int e    = gid >> 3;
    int part = gid & 7;
    if (e >= N_EDGES) return;
    int s = src[e];
    int d = dst[e];
    const float4 v = *(const float4*)(h1 + (long)s * HIDDEN + part * 4);
    float* ad = agg2 + (long)d * HIDDEN + part * 4;
    atomicAdd(ad + 0, v.x);
    atomicAdd(ad + 1, v.y);
    atomicAdd(ad + 2, v.z);
    atomicAdd(ad + 3, v.w);
}

// -------------------- layer 2: h2 = ((agg2+h1)/(deg+1)) @ W2 + b2
// one wave per 16-node tile; M=16, N=64 (4 tiles), K=32 (8 k-steps)
__global__ void layer2_wmma(const float* __restrict__ h1,
                            const float* __restrict__ agg2,
                            const float* __restrict__ deg,
                            const float* __restrict__ W2,
                            const float* __restrict__ bias2,
                            float* __restrict__ h2) {
    const int lane = threadIdx.x & 31;
    const int wave = threadIdx.x >> 5;
    const int tile = blockIdx.x * (blockDim.x >> 5) + wave;
    const int NT = N_NODES / 16;
    if (tile >= NT) return;                   // wave-uniform

    const int row  = lane & 15;
    const int kh   = lane >> 4;
    const int node = tile * 16 + row;
    const float inv = 1.0f / (deg[node] + 1.0f);
    const v2f* __restrict__ ap = (const v2f*)(agg2 + (long)node * HIDDEN);
    const v2f* __restrict__ hp = (const v2f*)(h1   + (long)node * HIDDEN);

    v8f acc[4] = {{}, {}, {}, {}};
#pragma unroll
    for (int kk = 0; kk < 8; ++kk) {
        const int k0 = kk * 4 + kh * 2;       // 0..30, always in range, even
        v2f a = (ap[k0 >> 1] + hp[k0 >> 1]) * inv;   // b64 vector loads
#pragma unroll
        for (int t = 0; t < 4; ++t) {
            v2f b;
            b.x = W2[(long)(k0    ) * EMB + t * 16 + row];
            b.y = W2[(long)(k0 + 1) * EMB + t * 16 + row];
            acc[t] = __builtin_amdgcn_wmma_f32_16x16x4_f32(false, a, false, b, (short)0, acc[t], false, false);
        }
    }

#pragma unroll
    for (int t = 0; t < 4; ++t) {
        const int n = t * 16 + row;
        const float bv = bias2[n];
#pragma unroll
        for (int r = 0; r < 8; ++r) {
            const int m = r + 8 * kh;
            h2[(long)(tile * 16 + m) * EMB + n] = acc[t][r] + bv;
        }
    }
}

// ---------------------------------------- per-graph mean: sums + counts
// graph_id is sorted, so a block's 4 consecutive nodes mostly share one
// graph: do a segmented pre-reduction in LDS, then one atomic per segment.
__global__ void pool_sum(const float* __restrict__ h2,
                         const int* __restrict__ graph_id,
                         float* __restrict__ out,
                         float* __restrict__ cnt) {
    __shared__ float sv[256];
    __shared__ int   sg[4];
    const int i    = blockIdx.x * 256 + threadIdx.x;   // node*64 + f  (exact: 100000*64 % 256 == 0)
    const int node = i >> 6;
    const int f    = i & 63;
    const int nl   = threadIdx.x >> 6;                 // node-local id 0..3

    const float v = h2[i];
    const int   g = graph_id[node];
    sv[threadIdx.x] = v;
    if (f == 0) sg[nl] = g;
    __syncthreads();

    const bool head = (nl == 0) || (sg[nl - 1] != g);  // start of a graph segment
    if (head) {
        float s  = v;
        int   len = 1;
#pragma unroll
        for (int j = 1; j < 4; ++j) {
            int k = nl + j;
            if (k < 4 && sg[k] == g) { s += sv[k * 64 + f]; ++len; }
            else break;
        }
        atomicAdd(&out[g * EMB + f], s);
        if (f == 0) atomicAdd(&cnt[g], (float)len);
    }
}

__global__ void pool_div(float* __restrict__ out, const float* __restrict__ cnt) {
    int i = blockIdx.x * blockDim.x + threadIdx.x;
    if (i >= N_GRAPHS * EMB) return;
    float c = cnt[i >> 6];
    out[i] = out[i] / fmaxf(c, 1.0f);
}

// ------------------------------------------------------------------ host
extern "C" void kernel_launch(void* const* d_in, const int* in_sizes, int n_in,
                              void* d_out, int out_size, void* d_ws, size_t ws_size,
                              hipStream_t stream) {
    const float* x        = (const float*)d_in[0];
    const int*   src      = (const int*)  d_in[1];
    const int*   dst      = (const int*)  d_in[2];
    const int*   graph_id = (const int*)  d_in[3];
    const float* W1       = (const float*)d_in[4];
    const float* b1       = (const float*)d_in[5];
    const float* W2       = (const float*)d_in[6];
    const float* b2       = (const float*)d_in[7];
    float* out = (float*)d_out;

    // workspace layout (floats): [deg | agg1 | agg2 | cnt | h1 | h2]  (~54 MB)
    float* ws   = (float*)d_ws;
    float* deg  = ws;                                   // 100000
    float* agg1 = deg  + N_NODES;                       // 600000
    float* agg2 = agg1 + (long)N_NODES * IN_FEATS;      // 3.2M
    float* cnt  = agg2 + (long)N_NODES * HIDDEN;        // 256
    float* h1   = cnt  + N_GRAPHS;                      // 3.2M
    float* h2   = h1   + (long)N_NODES * HIDDEN;        // 6.4M
    const int n_accum = N_NODES + N_NODES * IN_FEATS + N_NODES * HIDDEN + N_GRAPHS;

    const int B = 256;

    // 1. zero atomic accumulators + output (atomics must restart clean每 call)
    zero_f32<<<(n_accum + B - 1) / B, B, 0, stream>>>(ws, n_accum);
    zero_f32<<<(N_GRAPHS * EMB + B - 1) / B, B, 0, stream>>>(out, N_GRAPHS * EMB);

    // 2. degree + layer-1 aggregation
    scatter1<<<(N_EDGES + B - 1) / B, B, 0, stream>>>(x, src, dst, deg, agg1);

    // 3. layer-1 dense (WMMA f32), 6250 tiles, 8 waves/block
    const int NT  = N_NODES / 16;
    const int wpb = B / 32;
    layer1_wmma<<<(NT + wpb - 1) / wpb, B, 0, stream>>>(x, agg1, deg, W1, b1, h1);

    // 4. layer-2 aggregation (8 threads per edge)
    scatter2<<<((N_EDGES * 8) + B - 1) / B, B, 0, stream>>>(h1, src, dst, agg2);

    // 5. layer-2 dense (WMMA f32)
    layer2_wmma<<<(NT + wpb - 1) / wpb, B, 0, stream>>>(h1, agg2, deg, W2, b2, h2);

    // 6. mean pooling (LDS segmented pre-reduction, then atomics)
    pool_sum<<<(N_NODES * EMB) / B, B, 0, stream>>>(h2, graph_id, out, cnt);
    pool_div<<<((N_GRAPHS * EMB) + B - 1) / B, B, 0, stream>>>(out, cnt);
}